// DynamicWeightedMSELoss_7713761264101
// MI455X (gfx1250) — compile-verified
//
#include <hip/hip_runtime.h>
#include <hip/hip_bf16.h>
#include <math.h>

// ---------------------------------------------------------------------------
// DynamicWeightedMSELoss, fused single pass at the HBM roofline.
//   pass 1 (histo_sse_kernel):
//     - CDNA5 async global->LDS staging (GLOBAL_LOAD_ASYNC_TO_LDS_B128,
//       ASYNCcnt), 4-stage per-wave ring, 3 stages issued ahead.
//     - per-sample bin + squared error -> per-wave, 8x sub-replicated LDS
//       histograms (ds_add_u32 / ds_add_f32, ~conflict-free).
//     - per-block flush via global f32 atomics into 40-float workspace.
//   pass 2 (finalize_kernel): 20-bin weight math + weighted sum -> d_out[0].
// Traffic: 2 * 2^24 * 4B = 134 MB read once => ~5.8 us at 23.3 TB/s.
// ---------------------------------------------------------------------------

#define NUM_BINS 20
#define BETA 0.9f
#define BLOCK_SIZE 256
#define WAVES_PER_BLOCK 8   // wave32: 256 threads = 8 waves
#define GRID_BLOCKS 2048
#define NSUB 8              // histogram sub-replication within a wave
#define PIPE 4              // async staging ring depth (issue 3 ahead)
// One "chunk" = 32 lanes * 4 floats = 128 floats per array.

#if __has_builtin(__builtin_amdgcn_global_load_async_to_lds_b128)
#define HAVE_ASYNC_LDS 1
#else
#define HAVE_ASYNC_LDS 0
#endif

#if __has_builtin(__builtin_amdgcn_s_wait_asynccnt)
#define WAIT_ASYNC(n) __builtin_amdgcn_s_wait_asynccnt(n)
#else
#define WAIT_ASYNC(n) asm volatile("s_wait_asynccnt %0" ::"n"(n) : "memory")
#endif

// The async-LDS builtin takes pointers to GCC-vector int4 in AS1 / AS3.
typedef int v4i __attribute__((vector_size(16)));
typedef __attribute__((address_space(1))) v4i gbl_v4i_t;
typedef __attribute__((address_space(3))) v4i lds_v4i_t;

// Generic LDS pointer -> AS3: low 32 bits of the generic (shared-aperture)
// address are the LDS byte offset, which is exactly an AS3 pointer value.
__device__ __forceinline__ lds_v4i_t* to_lds(void* p) {
  return (lds_v4i_t*)(unsigned int)(unsigned long long)p;
}
__device__ __forceinline__ gbl_v4i_t* to_gbl(const void* p) {
  return (gbl_v4i_t*)(unsigned long long)p;
}

__global__ void __launch_bounds__(BLOCK_SIZE)
histo_sse_kernel(const float* __restrict__ pred,
                 const float* __restrict__ target,
                 float* __restrict__ ws_cnt,   // [NUM_BINS]
                 float* __restrict__ ws_sse,   // [NUM_BINS]
                 int n) {
  // [wave][bin][sub]: same-address atomic collisions only among the 4 lanes
  // sharing (lane & 7); [bin][sub] layout spreads across LDS banks.
  __shared__ unsigned s_cnt[WAVES_PER_BLOCK][NUM_BINS][NSUB];
  __shared__ float s_sse[WAVES_PER_BLOCK][NUM_BINS][NSUB];
#if HAVE_ASYNC_LDS
  // [wave][ring-stage][pred|target][lane] : 8*4*2*32*16B = 32 KB
  __shared__ float4 s_stage[WAVES_PER_BLOCK][PIPE][2][32];
#endif

  const int tid = threadIdx.x;
  const int wid = tid >> 5;
  const int lane = tid & 31;
  const int sub = lane & (NSUB - 1);

  for (int i = tid; i < WAVES_PER_BLOCK * NUM_BINS * NSUB; i += BLOCK_SIZE) {
    (&s_cnt[0][0][0])[i] = 0u;
    (&s_sse[0][0][0])[i] = 0.0f;
  }
  __syncthreads();

  auto process1 = [&](float p, float t) {
    const float d = p - t;
    int b = (int)(t * (float)NUM_BINS);          // floor for t >= 0
    b = b < 0 ? 0 : (b > NUM_BINS - 1 ? NUM_BINS - 1 : b);
    __hip_atomic_fetch_add(&s_cnt[wid][b][sub], 1u, __ATOMIC_RELAXED,
                           __HIP_MEMORY_SCOPE_WORKGROUP);   // ds_add_u32
    __hip_atomic_fetch_add(&s_sse[wid][b][sub], d * d, __ATOMIC_RELAXED,
                           __HIP_MEMORY_SCOPE_WORKGROUP);   // ds_add_f32
  };
  auto process4 = [&](float4 p4, float4 t4) {
    process1(p4.x, t4.x);
    process1(p4.y, t4.y);
    process1(p4.z, t4.z);
    process1(p4.w, t4.w);
  };

  const long long nChunks = (long long)(n >> 7);  // 128 floats per chunk
  const long long waveId = (long long)blockIdx.x * WAVES_PER_BLOCK + wid;
  const long long waveStride = (long long)gridDim.x * WAVES_PER_BLOCK;

#if HAVE_ASYNC_LDS
  // Per-wave async staging ring: each wave stages and consumes its own tiles,
  // so only per-wave ASYNCcnt waits are needed (no cross-wave barriers).
  auto issue = [&](int slot, long long chunk) {
    const float* gp = pred + (chunk << 7) + (lane << 2);
    const float* gt = target + (chunk << 7) + (lane << 2);
    __builtin_amdgcn_global_load_async_to_lds_b128(
        to_gbl(gp), to_lds(&s_stage[wid][slot][0][lane]), 0, 0);
    __builtin_amdgcn_global_load_async_to_lds_b128(
        to_gbl(gt), to_lds(&s_stage[wid][slot][1][lane]), 0, 0);
  };

  long long c = waveId;
  long long rem =
      (nChunks > waveId) ? ((nChunks - waveId + waveStride - 1) / waveStride)
                         : 0;  // chunks this wave still has to consume
  // Prologue: fill up to PIPE-1 stages ahead.
#pragma unroll
  for (int j = 0; j < PIPE - 1; ++j) {
    if (j < rem) issue(j, c + (long long)j * waveStride);
  }
  unsigned k = 0;  // consume counter; slot = k & (PIPE-1)
  while (rem > 0) {
    if (rem > PIPE - 1) issue((k + PIPE - 1) & (PIPE - 1),
                              c + (long long)(PIPE - 1) * waveStride);
    // Stages in flight right now = min(rem, PIPE); wait until only the
    // younger (inflight-1) stages (2 async ops each) remain outstanding.
    const int infl = rem >= PIPE ? PIPE : (int)rem;
    switch (infl) {
      case 4: WAIT_ASYNC(6); break;
      case 3: WAIT_ASYNC(4); break;
      case 2: WAIT_ASYNC(2); break;
      default: WAIT_ASYNC(0); break;
    }
    asm volatile("" ::: "memory");  // keep DS reads below the wait
    const int slot = k & (PIPE - 1);
    const float4 p4 = s_stage[wid][slot][0][lane];
    const float4 t4 = s_stage[wid][slot][1][lane];
    process4(p4, t4);
    ++k;
    c += waveStride;
    --rem;
  }
  const long long done = nChunks << 7;
#else
  // Fallback: plain grid-stride B128 loads (still HBM-roofline for streaming).
  {
    const float4* p4p = (const float4*)pred;
    const float4* t4p = (const float4*)target;
    const long long n4 = (long long)(n >> 2);
    const long long stride = (long long)gridDim.x * BLOCK_SIZE;
    for (long long i = (long long)blockIdx.x * BLOCK_SIZE + tid; i < n4;
         i += stride) {
      process4(p4p[i], t4p[i]);
    }
  }
  const long long done = (long long)(n >> 2) << 2;
#endif

  // Tail (n not a multiple of the vector width): handled by block 0.
  if (blockIdx.x == 0) {
    for (long long i = done + tid; i < (long long)n; i += BLOCK_SIZE) {
      process1(pred[i], target[i]);
    }
  }

  __syncthreads();
  // Combine the 8x8 wave/sub-private histograms; 20 lanes -> global atomics.
  if (tid < NUM_BINS) {
    unsigned ctot = 0;
    float stot = 0.0f;
    for (int w = 0; w < WAVES_PER_BLOCK; ++w) {
      for (int s = 0; s < NSUB; ++s) {
        ctot += s_cnt[w][tid][s];
        stot += s_sse[w][tid][s];
      }
    }
    if (ctot) {
      __hip_atomic_fetch_add(&ws_cnt[tid], (float)ctot, __ATOMIC_RELAXED,
                             __HIP_MEMORY_SCOPE_AGENT);
      __hip_atomic_fetch_add(&ws_sse[tid], stot, __ATOMIC_RELAXED,
                             __HIP_MEMORY_SCOPE_AGENT);
    }
  }
}

__global__ void init_ws_kernel(float* ws) {
  const int i = threadIdx.x;
  if (i < 2 * NUM_BINS) ws[i] = 0.0f;
}

__global__ void finalize_kernel(const float* __restrict__ ws_cnt,
                                const float* __restrict__ ws_sse,
                                float* __restrict__ out, float inv_n) {
  if (threadIdx.x == 0) {
    float w[NUM_BINS];
    float s = 0.0f;
    for (int b = 0; b < NUM_BINS; ++b) {
      const float c = fmaxf(ws_cnt[b], 1.0f);
      w[b] = powf(c, -BETA);
      s += w[b];
    }
    float acc = 0.0f;
    for (int b = 0; b < NUM_BINS; ++b) {
      float wb = (s > 0.0f) ? (w[b] / s * (float)NUM_BINS) : w[b];
      wb = fmaxf(wb, 1.0f);  // MIN_WEIGHT
      acc += wb * ws_sse[b];
    }
    *out = acc * inv_n;
  }
}

extern "C" void kernel_launch(void* const* d_in, const int* in_sizes, int n_in,
                              void* d_out, int out_size, void* d_ws,
                              size_t ws_size, hipStream_t stream) {
  const float* pred = (const float*)d_in[0];
  const float* target = (const float*)d_in[1];
  const int n = in_sizes[0];

  float* ws_cnt = (float*)d_ws;            // 20 floats
  float* ws_sse = ws_cnt + NUM_BINS;       // 20 floats

  hipLaunchKernelGGL(init_ws_kernel, dim3(1), dim3(64), 0, stream,
                     (float*)d_ws);
  hipLaunchKernelGGL(histo_sse_kernel, dim3(GRID_BLOCKS), dim3(BLOCK_SIZE), 0,
                     stream, pred, target, ws_cnt, ws_sse, n);
  hipLaunchKernelGGL(finalize_kernel, dim3(1), dim3(32), 0, stream, ws_cnt,
                     ws_sse, (float*)d_out, 1.0f / (float)n);
}